// PostProcessor_49469433315559
// MI455X (gfx1250) — compile-verified
//
#include <hip/hip_runtime.h>

typedef unsigned int u32;
typedef __attribute__((ext_vector_type(2))) float v2f;
typedef __attribute__((ext_vector_type(8))) float v8f;

#define B_ 8
#define N_ 1024
#define C_ 151
#define R_ 131072
#define P_ 51
#define TOTREL (B_*R_)          /* 1048576 */
#define NBLK 128                /* radix blocks per batch: 131072/1024 */

__device__ __forceinline__ float wredMax(float v){
  #pragma unroll
  for (int off = 16; off; off >>= 1) v = fmaxf(v, __shfl_xor(v, off, 32));
  return v;
}
__device__ __forceinline__ float wredSum(float v){
  #pragma unroll
  for (int off = 16; off; off >>= 1) v += __shfl_xor(v, off, 32);
  return v;
}
__device__ __forceinline__ void wredArgmax(float& v, int& c){
  #pragma unroll
  for (int off = 16; off; off >>= 1){
    float ov = __shfl_xor(v, off, 32);
    int   oc = __shfl_xor(c, off, 32);
    if (ov > v || (ov == v && oc < c)){ v = ov; c = oc; }
  }
}
// Monotonic key: ascending radix sort of key == descending sort of float (stable).
__device__ __forceinline__ u32 fkey_desc(float f){
  u32 u = __float_as_uint(f);
  u32 k = (u & 0x80000000u) ? ~u : (u | 0x80000000u);
  return ~k;
}

// ---------------- Kernel 1: object softmax stats (wave per row) ----------------
__global__ void __launch_bounds__(256) obj_kernel(const float* __restrict__ obj_logit,
                                                  float* __restrict__ out_pred,
                                                  float* __restrict__ out_score){
  int wave = threadIdx.x >> 5, lane = threadIdx.x & 31;
  int row = blockIdx.x * 8 + wave;                       // 8192 rows total
  const float* p = obj_logit + (size_t)row * C_;
  float m = -3.4e38f;
  for (int c = lane; c < C_; c += 32) m = fmaxf(m, p[c]);
  m = wredMax(m);
  float s = 0.f, bv = -3.4e38f; int bc = 1 << 30;
  for (int c = lane; c < C_; c += 32){
    float v = p[c];
    s += expf(v - m);
    if (c >= 1 && (v > bv || (v == bv && c < bc))){ bv = v; bc = c; }
  }
  s = wredSum(s);
  wredArgmax(bv, bc);
  if (lane == 0){
    out_pred[row]  = (float)bc;            // argmax(fg)+1 == original class index
    out_score[row] = expf(bv - m) / s;
  }
}

// -------- Kernel 2: rel softmax sums via WMMA + triple-score keys (32 rows/wave) --------
__global__ void __launch_bounds__(256) rel_key_kernel(const float* __restrict__ rel_logit,
                                                      const int*   __restrict__ pair,
                                                      const float* __restrict__ obj_score,
                                                      u32* __restrict__ keys,
                                                      u32* __restrict__ idxs){
  __shared__ float tile[8][32 * 52];     // 52 = 51 + zero-pad col -> 13 K-chunks of 4
  __shared__ float sums[8][32];
  int t = threadIdx.x, wave = t >> 5, lane = t & 31;
  long long rowBase = (long long)blockIdx.x * 256 + (long long)wave * 32;
  float* tw = tile[wave];
  const float* src = rel_logit + rowBase * P_;
  for (int i = lane; i < 32 * P_; i += 32){              // coalesced span load
    int r = i / P_, c = i - r * P_;
    tw[r * 52 + c] = src[i];
  }
  tw[lane * 52 + 51] = 0.f;
  __syncthreads();

  const float* myr = tw + lane * 52;                     // each lane owns one row
  float m = -3.4e38f;
  for (int c = 0; c < P_; ++c) m = fmaxf(m, myr[c]);
  float bv = -3.4e38f; int bc = 1;
  for (int c = 1; c < P_; ++c){ float v = myr[c]; if (v > bv){ bv = v; bc = c; } }
  float* myw = tw + lane * 52;
  for (int c = 0; c < P_; ++c) myw[c] = expf(myr[c] - m);
  __syncthreads();

  // Row sums of exp-tile via V_WMMA_F32_16X16X4_F32 with B = ones.
  v8f acc0 = {0.f,0.f,0.f,0.f,0.f,0.f,0.f,0.f};          // rows 0..15
  v8f acc1 = {0.f,0.f,0.f,0.f,0.f,0.f,0.f,0.f};          // rows 16..31
  v2f ones; ones.x = 1.f; ones.y = 1.f;
  int mrow = lane & 15;
  int kb   = (lane >> 4) * 2;                            // A layout: lanes 0-15 K={0,1}, 16-31 K={2,3}
  #pragma unroll
  for (int it = 0; it < 13; ++it){
    int k = it * 4 + kb;
    v2f a0, a1;
    a0.x = tw[mrow * 52 + k];        a0.y = tw[mrow * 52 + k + 1];
    a1.x = tw[(mrow + 16) * 52 + k]; a1.y = tw[(mrow + 16) * 52 + k + 1];
    acc0 = __builtin_amdgcn_wmma_f32_16x16x4_f32(false, a0, false, ones, (short)0, acc0, false, false);
    acc1 = __builtin_amdgcn_wmma_f32_16x16x4_f32(false, a1, false, ones, (short)0, acc1, false, false);
  }
  // D layout: VGPR j holds M=j (lanes 0-15) / M=j+8 (lanes 16-31); all N columns equal the row sum.
  if (lane == 0){
    #pragma unroll
    for (int j = 0; j < 8; ++j){ sums[wave][j] = acc0[j]; sums[wave][16 + j] = acc1[j]; }
  } else if (lane == 16){
    #pragma unroll
    for (int j = 0; j < 8; ++j){ sums[wave][8 + j] = acc0[j]; sums[wave][24 + j] = acc1[j]; }
  }
  __syncthreads();

  float s = sums[wave][lane];
  float relsc = expf(bv - m) / s;                        // max fg prob
  long long grow = rowBase + lane;
  int b = (int)(grow >> 17);
  int r = (int)(grow & (R_ - 1));
  int p0 = pair[grow * 2 + 0];
  int p1 = pair[grow * 2 + 1];
  float trip = relsc * obj_score[b * N_ + p0] * obj_score[b * N_ + p1];
  keys[grow] = fkey_desc(trip);
  idxs[grow] = (u32)r;
}

// ---------------- Radix sort pass kernels (stable LSD, per batch) ----------------
__global__ void __launch_bounds__(256) hist_kernel(const u32* __restrict__ keys,
                                                   u32* __restrict__ counts, int shift){
  __shared__ u32 h[256];
  int t = threadIdx.x;
  h[t] = 0;
  __syncthreads();
  int b = blockIdx.y, blk = blockIdx.x;
  size_t base = (size_t)b * R_ + (size_t)blk * 1024;
  #pragma unroll
  for (int j = 0; j < 4; ++j){
    u32 d = (keys[base + j * 256 + t] >> shift) & 255u;
    atomicAdd(&h[d], 1u);
  }
  __syncthreads();
  counts[(size_t)b * (256 * NBLK) + t * NBLK + blk] = h[t];   // digit-major
}

__global__ void __launch_bounds__(1024) scan_kernel(u32* __restrict__ counts){
  __shared__ u32 sc[1024];
  int t = threadIdx.x;
  u32* arr = counts + (size_t)blockIdx.x * (256 * NBLK);      // 32768 entries per batch
  u32 loc[32];
  u32 run = 0;
  int base = t * 32;
  #pragma unroll
  for (int i = 0; i < 32; ++i){ u32 x = arr[base + i]; loc[i] = run; run += x; }
  sc[t] = run;
  __syncthreads();
  for (int off = 1; off < 1024; off <<= 1){
    u32 v = (t >= off) ? sc[t - off] : 0u;
    __syncthreads();
    sc[t] += v;
    __syncthreads();
  }
  u32 excl = sc[t] - run;
  #pragma unroll
  for (int i = 0; i < 32; ++i) arr[base + i] = loc[i] + excl;
}

__global__ void __launch_bounds__(256) scatter_kernel(const u32* __restrict__ keysIn,
                                                      const u32* __restrict__ idxIn,
                                                      u32* __restrict__ keysOut,
                                                      u32* __restrict__ idxOut,
                                                      const u32* __restrict__ counts, int shift){
  __shared__ u32 running[256];
  __shared__ u32 wcnt[8][256];
  int t = threadIdx.x, wave = t >> 5, lane = t & 31;
  int b = blockIdx.y, blk = blockIdx.x;
  size_t base = (size_t)b * R_ + (size_t)blk * 1024;
  const u32* scn = counts + (size_t)b * (256 * NBLK);
  u32 k[4], v[4];
  #pragma unroll
  for (int j = 0; j < 4; ++j){                 // item order: j*256 + t  (== original order)
    k[j] = keysIn[base + j * 256 + t];
    v[j] = idxIn [base + j * 256 + t];
  }
  running[t] = 0;
  for (int j = 0; j < 4; ++j){
    #pragma unroll
    for (int w = 0; w < 8; ++w) wcnt[w][t] = 0;
    __syncthreads();
    u32 d = (k[j] >> shift) & 255u;
    u32 mask = 0xFFFFFFFFu;                    // wave32 ballot peer match (stable, lane-ordered)
    #pragma unroll
    for (int bit = 0; bit < 8; ++bit){
      u32 bal = (u32)__ballot((d >> bit) & 1);
      mask &= ((d >> bit) & 1) ? bal : ~bal;
    }
    int lrank  = __popc(mask & ((1u << lane) - 1u));
    int leader = __builtin_ctz(mask);
    if (lane == leader) wcnt[wave][d] = (u32)__popc(mask);
    __syncthreads();
    u32 cross = 0;
    for (int w = 0; w < wave; ++w) cross += wcnt[w][d];
    u32 pos = scn[d * NBLK + blk] + running[d] + cross + (u32)lrank;
    keysOut[(size_t)b * R_ + pos] = k[j];
    idxOut [(size_t)b * R_ + pos] = v[j];
    __syncthreads();
    u32 tot = 0;
    #pragma unroll
    for (int w = 0; w < 8; ++w) tot += wcnt[w][t];
    running[t] += tot;
    __syncthreads();
  }
}

// ---------------- Kernel 4: sorted gather + softmax recompute (wave per rank) ----------------
__global__ void __launch_bounds__(256) gather_kernel(const float* __restrict__ rel_logit,
                                                     const int*   __restrict__ pair,
                                                     const u32*   __restrict__ idxSorted,
                                                     float* __restrict__ out_pair,
                                                     float* __restrict__ out_prob,
                                                     float* __restrict__ out_label){
  int wave = threadIdx.x >> 5, lane = threadIdx.x & 31;
  int gw = blockIdx.x * 8 + wave;              // 0 .. TOTREL-1
  int b = gw >> 17;
  u32 r = idxSorted[gw];
  size_t srow = ((size_t)b << 17) + r;
  const float* row = rel_logit + srow * P_;
  float v0 = (lane < P_)      ? row[lane]      : -3.4e38f;
  float v1 = (lane + 32 < P_) ? row[lane + 32] : -3.4e38f;
  float m = wredMax(fmaxf(v0, v1));
  float e0 = (lane < P_)      ? expf(v0 - m) : 0.f;
  float e1 = (lane + 32 < P_) ? expf(v1 - m) : 0.f;
  float s = wredSum(e0 + e1);
  float bv = -3.4e38f; int bc = 1 << 30;
  if (lane >= 1 && lane < P_){ bv = v0; bc = lane; }
  if (lane + 32 < P_ && v1 > bv){ bv = v1; bc = lane + 32; }
  wredArgmax(bv, bc);
  float* op = out_prob + (size_t)gw * P_;
  if (lane < P_)      op[lane]      = e0 / s;
  if (lane + 32 < P_) op[lane + 32] = e1 / s;
  if (lane == 0){
    out_pair[(size_t)gw * 2 + 0] = (float)pair[srow * 2 + 0];
    out_pair[(size_t)gw * 2 + 1] = (float)pair[srow * 2 + 1];
    out_label[gw] = (float)bc;
  }
}

extern "C" void kernel_launch(void* const* d_in, const int* in_sizes, int n_in,
                              void* d_out, int out_size, void* d_ws, size_t ws_size,
                              hipStream_t stream){
  const float* obj_logit = (const float*)d_in[0];
  const float* rel_logit = (const float*)d_in[1];
  const int*   pair      = (const int*)d_in[2];

  float* out       = (float*)d_out;
  float* out_pred  = out;                              // B*N
  float* out_score = out_pred + B_ * N_;               // B*N
  float* out_pair  = out_score + B_ * N_;              // B*R*2
  float* out_prob  = out_pair + 2ll * TOTREL;          // B*R*P
  float* out_label = out_prob + (long long)P_ * TOTREL;// B*R

  u32* keysA  = (u32*)d_ws;                            // TOTREL
  u32* idxA   = keysA + TOTREL;
  u32* keysB  = idxA + TOTREL;
  u32* idxB   = keysB + TOTREL;
  u32* counts = idxB + TOTREL;                         // B * 256 * NBLK

  obj_kernel<<<1024, 256, 0, stream>>>(obj_logit, out_pred, out_score);
  rel_key_kernel<<<4096, 256, 0, stream>>>(rel_logit, pair, out_score, keysA, idxA);

  u32 *ks = keysA, *vs = idxA, *kd = keysB, *vd = idxB;
  for (int pass = 0; pass < 4; ++pass){
    hist_kernel<<<dim3(NBLK, B_), 256, 0, stream>>>(ks, counts, pass * 8);
    scan_kernel<<<B_, 1024, 0, stream>>>(counts);
    scatter_kernel<<<dim3(NBLK, B_), 256, 0, stream>>>(ks, vs, kd, vd, counts, pass * 8);
    u32* tk = ks; ks = kd; kd = tk;
    u32* tv = vs; vs = vd; vd = tv;
  }
  // After 4 passes, sorted key/idx live back in keysA/idxA (== ks/vs).
  gather_kernel<<<TOTREL / 8, 256, 0, stream>>>(rel_logit, pair, vs, out_pair, out_prob, out_label);
}